// CapsuleNetwork_90537910600110
// MI455X (gfx1250) — compile-verified
//
#include <hip/hip_runtime.h>
#include <math.h>

typedef __attribute__((ext_vector_type(2))) float v2f;
typedef __attribute__((ext_vector_type(8))) float v8f;

#define R_CAPS 128
#define D_BERT 768
#define NOUT   2
#define NATOM  16
#define NK     32                 // NOUT*NATOM
#define BATCH  512
#define KK_STEPS (D_BERT / 4)     // 192 K-steps of 4 for f32 WMMA
#define LW_ELEMS (KK_STEPS * 2 * 32 * 2)   // 24576 floats = 96KB swizzled W slab

// ---------------------------------------------------------------------------
// Kernel 1: P[b,r,k] = sum_d in[b,r,d] * W[r,d,k]  via V_WMMA_F32_16X16X4_F32
// grid = (128 r, 2 batch-tiles), block = 256 (8 waves). Each wave: 32x32 tile.
// ---------------------------------------------------------------------------
__global__ __launch_bounds__(256) void capsule_proj_wmma(
    const float* __restrict__ in,   // [512,128,768]
    const float* __restrict__ W,    // [128,768,32]
    float* __restrict__ P)          // [512,128,32]
{
  extern __shared__ float lw[];     // LW_ELEMS floats, WMMA-B swizzled
  const int r  = blockIdx.x;
  const int bt = blockIdx.y;

  // --- cooperative load of W[r] with swizzle into B-fragment layout --------
  // B 4x16 f32 fragment assumption (from ISA 8-bit B pattern):
  //   VGPR j, lane-half h -> K = 2*j + h, N = lane&15.
  // LDS element for (kk, nt, lane, pair): pair=j, lane encodes (K&1, N).
  const float* Wr = W + (size_t)r * (D_BERT * NK);
  for (int i = threadIdx.x; i < LW_ELEMS / 4; i += 256) {
    const float4 v = reinterpret_cast<const float4*>(Wr)[i];
    const float vv[4] = {v.x, v.y, v.z, v.w};
#pragma unroll
    for (int j = 0; j < 4; ++j) {
      const int f    = i * 4 + j;
      const int d    = f >> 5;          // row (0..767)
      const int n    = f & 31;          // col (0..31)
      const int kk   = d >> 2;
      const int koff = d & 3;
      const int lane = ((koff & 1) << 4) | (n & 15);
      const int nt   = n >> 4;
      const int pair = koff >> 1;
      lw[((kk * 2 + nt) * 32 + lane) * 2 + pair] = vv[j];
    }
  }
  __syncthreads();

  const int  wave = threadIdx.x >> 5;
  const int  lane = threadIdx.x & 31;
  const int  lh   = lane & 15;
  const bool hi   = lane >= 16;
  const int  m0   = bt * 256 + wave * 32;   // 32 batch rows per wave

  const float* arow0 = in + ((size_t)(m0 + lh) * R_CAPS + r) * D_BERT;
  const float* arow1 = arow0 + (size_t)16 * R_CAPS * D_BERT;
  const float* lwp   = lw + lane * 2;

  v8f c00 = {}, c01 = {}, c10 = {}, c11 = {};

#pragma unroll 4
  for (int kk = 0; kk < KK_STEPS; ++kk) {
    // A fragments: lanes 0-15 hold (K0,K1), lanes 16-31 hold (K2,K3), M=lane&15
    const float4 av0 = *reinterpret_cast<const float4*>(arow0 + kk * 4);
    const float4 av1 = *reinterpret_cast<const float4*>(arow1 + kk * 4);
    v2f a0, a1;
    a0.x = hi ? av0.z : av0.x;  a0.y = hi ? av0.w : av0.y;
    a1.x = hi ? av1.z : av1.x;  a1.y = hi ? av1.w : av1.y;
    // B fragments: single b64 each thanks to the LDS swizzle
    const v2f b0 = *reinterpret_cast<const v2f*>(lwp + kk * 128);
    const v2f b1 = *reinterpret_cast<const v2f*>(lwp + kk * 128 + 64);
    c00 = __builtin_amdgcn_wmma_f32_16x16x4_f32(false, a0, false, b0, (short)0, c00, false, false);
    c01 = __builtin_amdgcn_wmma_f32_16x16x4_f32(false, a0, false, b1, (short)0, c01, false, false);
    c10 = __builtin_amdgcn_wmma_f32_16x16x4_f32(false, a1, false, b0, (short)0, c10, false, false);
    c11 = __builtin_amdgcn_wmma_f32_16x16x4_f32(false, a1, false, b1, (short)0, c11, false, false);
  }

  // C/D layout: VGPR j, lanes0-15 -> M=j, lanes16-31 -> M=j+8, N=lane&15
  const int mo = hi ? 8 : 0;
#pragma unroll
  for (int j = 0; j < 8; ++j) {
    const size_t base0 = ((size_t)(m0 + mo + j)      * R_CAPS + r) * NK;
    const size_t base1 = ((size_t)(m0 + 16 + mo + j) * R_CAPS + r) * NK;
    P[base0 + lh]      = c00[j];
    P[base0 + 16 + lh] = c01[j];
    P[base1 + lh]      = c10[j];
    P[base1 + 16 + lh] = c11[j];
  }
}

// ---------------------------------------------------------------------------
// Kernel 2: dynamic routing (3 iters) + outputs + per-b margin cost
// grid = 512 (one block per b), block = 128 (one thread per capsule r)
// ---------------------------------------------------------------------------
__global__ __launch_bounds__(128) void capsule_route(
    const float* __restrict__ P,            // [512,128,32]
    const long long* __restrict__ labels,   // int64 [512]
    float* __restrict__ out_logits,         // [512,2]
    float* __restrict__ out_logitB,         // [512,128,2]
    float* __restrict__ out_Rw,             // [512,128,2]
    float* __restrict__ cost_ws)            // [512]
{
  const int b = blockIdx.x;
  const int r = threadIdx.x;

  __shared__ float S[NOUT * NATOM];
  __shared__ float sc[NOUT];
  __shared__ float lg[NOUT];

  float p[NOUT * NATOM];
  const float* Pr = P + ((size_t)b * R_CAPS + r) * NK;
#pragma unroll
  for (int i = 0; i < 8; ++i) {
    const float4 v = reinterpret_cast<const float4*>(Pr)[i];
    p[i * 4 + 0] = v.x; p[i * 4 + 1] = v.y;
    p[i * 4 + 2] = v.z; p[i * 4 + 3] = v.w;
  }

  float l0 = 0.f, l1 = 0.f, rw0 = 0.f, rw1 = 0.f;
  for (int it = 0; it < 3; ++it) {
    // softmax over the 2 output capsules
    const float m  = fmaxf(l0, l1);
    const float e0 = __expf(l0 - m), e1 = __expf(l1 - m);
    const float inv = 1.0f / (e0 + e1);
    rw0 = e0 * inv; rw1 = e1 * inv;

    if (threadIdx.x < NOUT * NATOM) S[threadIdx.x] = 0.f;
    __syncthreads();
#pragma unroll
    for (int a = 0; a < NATOM; ++a) {          // ds_add_f32 reduction over r
      atomicAdd(&S[a],         rw0 * p[a]);
      atomicAdd(&S[NATOM + a], rw1 * p[NATOM + a]);
    }
    __syncthreads();

    if (threadIdx.x < NOUT) {                  // squash scale per output
      const int o = threadIdx.x;
      float ns = 0.f;
#pragma unroll
      for (int a = 0; a < NATOM; ++a) { const float s = S[o * NATOM + a]; ns += s * s; }
      const float norm = sqrtf(ns);
      sc[o] = norm / (1.f + ns);               // V = S * sc
      lg[o] = ns   / (1.f + ns);               // ||V|| = norm * sc
    }
    __syncthreads();

    float ag0 = 0.f, ag1 = 0.f;                // agreement: sum_a P*V
#pragma unroll
    for (int a = 0; a < NATOM; ++a) {
      ag0 += p[a] * S[a];
      ag1 += p[NATOM + a] * S[NATOM + a];
    }
    l0 += ag0 * sc[0];
    l1 += ag1 * sc[1];
    __syncthreads();
  }

  const size_t ro = ((size_t)b * R_CAPS + r) * NOUT;
  out_logitB[ro + 0] = l0;  out_logitB[ro + 1] = l1;
  out_Rw[ro + 0]     = rw0; out_Rw[ro + 1]     = rw1;

  if (r == 0) {
    const float g0 = lg[0], g1 = lg[1];
    out_logits[b * 2 + 0] = g0;
    out_logits[b * 2 + 1] = g1;
    const int lab = (int)labels[b];
    float cost = 0.f;
    {
      const float on  = (lab == 0) ? 1.f : 0.f;
      const float pos = on * ((g0 < 0.95f) ? (0.95f - g0) * (0.95f - g0) : 0.f);
      const float neg = (1.f - on) * ((g0 > 0.05f) ? (g0 - 0.05f) * (g0 - 0.05f) : 0.f);
      cost += 0.5f * pos + 0.25f * neg;
    }
    {
      const float on  = (lab == 1) ? 1.f : 0.f;
      const float pos = on * ((g1 < 0.95f) ? (0.95f - g1) * (0.95f - g1) : 0.f);
      const float neg = (1.f - on) * ((g1 > 0.05f) ? (g1 - 0.05f) * (g1 - 0.05f) : 0.f);
      cost += 0.5f * pos + 0.25f * neg;
    }
    cost_ws[b] = cost;
  }
}

// ---------------------------------------------------------------------------
// Kernel 3: deterministic fixed-order reduction of 512 costs -> mean loss
// ---------------------------------------------------------------------------
__global__ __launch_bounds__(128) void reduce_loss(
    const float* __restrict__ cost, float* __restrict__ loss_out)
{
  __shared__ float red[128];
  const int t = threadIdx.x;
  red[t] = cost[t] + cost[t + 128] + cost[t + 256] + cost[t + 384];
  __syncthreads();
  for (int off = 64; off > 0; off >>= 1) {
    if (t < off) red[t] += red[t + off];
    __syncthreads();
  }
  if (t == 0) *loss_out = red[0] * (1.0f / 512.0f);
}

// ---------------------------------------------------------------------------
extern "C" void kernel_launch(void* const* d_in, const int* in_sizes, int n_in,
                              void* d_out, int out_size, void* d_ws, size_t ws_size,
                              hipStream_t stream) {
  (void)in_sizes; (void)n_in; (void)out_size; (void)ws_size;
  const float*     in     = (const float*)d_in[0];      // [512,128,768] f32
  const long long* labels = (const long long*)d_in[1];  // [512] int64
  const float*     W      = (const float*)d_in[2];      // [128,768,32] f32

  float* out        = (float*)d_out;
  float* outLogits  = out;                        // 1024
  float* outLoss    = out + 1024;                 // 1
  float* outLogitB  = out + 1025;                 // 131072
  float* outRw      = out + 1025 + 131072;        // 131072

  float* P       = (float*)d_ws;                            // 512*128*32 f32
  float* cost_ws = P + (size_t)BATCH * R_CAPS * NK;         // 512 f32

  capsule_proj_wmma<<<dim3(R_CAPS, BATCH / 256), 256,
                      LW_ELEMS * sizeof(float), stream>>>(in, W, P);
  capsule_route<<<BATCH, R_CAPS, 0, stream>>>(P, labels, outLogits, outLogitB,
                                              outRw, cost_ws);
  reduce_loss<<<1, 128, 0, stream>>>(cost_ws, outLoss);
}